// DGCNN_81466939670829
// MI455X (gfx1250) — compile-verified
//
#include <hip/hip_runtime.h>
#include <hip/hip_bf16.h>
#include <math.h>

#define BB 8
#define NN 2048
#define KK 20
#define BN_EPS 1e-5f
#define XC_LD 512

typedef float v2f __attribute__((ext_vector_type(2)));
typedef float v4f __attribute__((ext_vector_type(4)));
typedef float v8f __attribute__((ext_vector_type(8)));

__device__ __forceinline__ v8f wmma4(v2f a, v2f b, v8f c) {
    return __builtin_amdgcn_wmma_f32_16x16x4_f32(false, a, false, b, (short)0, c, false, false);
}

// ---------------------------------------------------------------- x (B,3,N) -> (B*N, 4) zero-padded
__global__ void transpose_x_kernel(const float* __restrict__ x, float* __restrict__ XT) {
    int t = blockIdx.x * blockDim.x + threadIdx.x;
    if (t >= BB * NN * 4) return;
    int c = t & 3;
    int n = (t >> 2) % NN;
    int b = t / (4 * NN);
    XT[t] = (c < 3) ? x[((size_t)b * 3 + c) * NN + n] : 0.f;
}

// ---------------------------------------------------------------- prep padded Wa / (Wb - Wa)
__global__ void wprep_kernel(const float* __restrict__ W, int cin, int cinp, int cout,
                             float* __restrict__ Wap, float* __restrict__ Wdp) {
    int t = blockIdx.x * blockDim.x + threadIdx.x;
    if (t >= cout * cinp) return;
    int o = t / cinp, c = t % cinp;
    float wa = (c < cin) ? W[(size_t)o * 2 * cin + c] : 0.f;
    float wb = (c < cin) ? W[(size_t)o * 2 * cin + cin + c] : 0.f;
    Wap[t] = wa;
    Wdp[t] = wb - wa;
}

// ---------------------------------------------------------------- |x|^2 per point
__global__ void sq_kernel(const float* __restrict__ X, int lda, int cin,
                          float* __restrict__ sq) {
    int t = blockIdx.x * blockDim.x + threadIdx.x;
    if (t >= BB * NN) return;
    const float* r = X + (size_t)t * lda;
    float s = 0.f;
    for (int c = 0; c < cin; ++c) s += r[c] * r[c];
    sq[t] = s;
}

// ---------------------------------------------------------------- WMMA tile core (no guards)
// A: rows at stride lda, B: rows at stride ldb, both K-major; K % 4 == 0, K >= 4.
// One wave accumulates a 16x64 tile. Software-pipelined: fragments for step k+1 are
// loaded before the WMMAs consuming step k, so dependent s_wait_loadcnt is partial
// and loads overlap the matrix pipe.
// Fragment layouts per ISA 7.12.2: A 16x4 f32 -> lane&15 = M, lane>>4 selects K pair;
// C/D v8f -> M = v + 8*(lane>>4), N = lane&15.
__device__ __forceinline__ void wmma_tile_core(const float* __restrict__ A, int lda,
                                               const float* __restrict__ B, int ldb,
                                               int K, int tileM, int tileN, int lane,
                                               v8f acc[4]) {
    const int half = lane >> 4;
    const int l16  = lane & 15;
    const float* ap  = A + (size_t)(tileM + l16) * lda + 2 * half;
    const float* bp0 = B + (size_t)(tileN + l16) * ldb + 2 * half;
    const float* bp1 = bp0 + (size_t)16 * ldb;
    const float* bp2 = bp0 + (size_t)32 * ldb;
    const float* bp3 = bp0 + (size_t)48 * ldb;

    v2f a  = *(const v2f*)(ap);
    v2f b0 = *(const v2f*)(bp0);
    v2f b1 = *(const v2f*)(bp1);
    v2f b2 = *(const v2f*)(bp2);
    v2f b3 = *(const v2f*)(bp3);
    for (int k0 = 4; k0 < K; k0 += 4) {
        v2f an  = *(const v2f*)(ap + k0);    // prefetch next step's fragments
        v2f bn0 = *(const v2f*)(bp0 + k0);
        v2f bn1 = *(const v2f*)(bp1 + k0);
        v2f bn2 = *(const v2f*)(bp2 + k0);
        v2f bn3 = *(const v2f*)(bp3 + k0);
        acc[0] = wmma4(a, b0, acc[0]);       // consume current step
        acc[1] = wmma4(a, b1, acc[1]);
        acc[2] = wmma4(a, b2, acc[2]);
        acc[3] = wmma4(a, b3, acc[3]);
        a = an; b0 = bn0; b1 = bn1; b2 = bn2; b3 = bn3;
    }
    acc[0] = wmma4(a, b0, acc[0]);
    acc[1] = wmma4(a, b1, acc[1]);
    acc[2] = wmma4(a, b2, acc[2]);
    acc[3] = wmma4(a, b3, acc[3]);
}

// ---------------------------------------------------------------- plain GEMM: C = A * B^T
__global__ void gemm_wmma_kernel(const float* __restrict__ A, int lda,
                                 const float* __restrict__ B, int ldb,
                                 float* __restrict__ C, int ldc, int K) {
    const int lane  = threadIdx.x & 31;
    const int half  = lane >> 4;
    const int l16   = lane & 15;
    const int tileM = blockIdx.y * 16;
    const int tileN = blockIdx.x * 64;
    v8f acc[4] = {};
    wmma_tile_core(A, lda, B, ldb, K, tileM, tileN, lane, acc);
#pragma unroll
    for (int t = 0; t < 4; ++t) {
        int n = tileN + 16 * t + l16;
#pragma unroll
        for (int v = 0; v < 8; ++v)
            C[(size_t)(tileM + v + 8 * half) * ldc + n] = acc[t][v];
    }
}

// ---------------------------------------------------------------- Gram scores for one batch:
// S[n,m] = 2 * x_n.x_m - |x_m|^2  (same ordering as reference pairwise distance)
__global__ void gram_kernel(const float* __restrict__ Xb, int lda, int K,
                            const float* __restrict__ sqb, float* __restrict__ S) {
    const int lane  = threadIdx.x & 31;
    const int half  = lane >> 4;
    const int l16   = lane & 15;
    const int tileM = blockIdx.y * 16;
    const int tileN = blockIdx.x * 64;
    v8f acc[4] = {};
    wmma_tile_core(Xb, lda, Xb, lda, K, tileM, tileN, lane, acc);
#pragma unroll
    for (int t = 0; t < 4; ++t) {
        int n = tileN + 16 * t + l16;
        float sm = sqb[n];
#pragma unroll
        for (int v = 0; v < 8; ++v)
            S[(size_t)(tileM + v + 8 * half) * NN + n] = 2.f * acc[t][v] - sm;
    }
}

// ---------------------------------------------------------------- streaming top-20 per row
// float4 streaming (rows are 16B aligned); stable insertion (strict >) matches
// jax.lax.top_k tie-breaking (lower index first).
__global__ void topk_kernel(const float* __restrict__ S, int* __restrict__ idxb) {
    int n = blockIdx.x * blockDim.x + threadIdx.x;
    if (n >= NN) return;
    const float* row = S + (size_t)n * NN;
    float tv[KK];
    int   ti[KK];
#pragma unroll
    for (int j = 0; j < KK; ++j) { tv[j] = -3.0e38f; ti[j] = 0; }
    for (int m4 = 0; m4 < NN; m4 += 4) {
        v4f s4 = *(const v4f*)(row + m4);
#pragma unroll
        for (int u = 0; u < 4; ++u) {
            float score = s4[u];
            if (score > tv[KK - 1]) {
                int pos = KK - 1;
#pragma unroll
                for (int j = KK - 2; j >= 0; --j)   // tv sorted desc -> condition monotone
                    if (score > tv[j]) { tv[j + 1] = tv[j]; ti[j + 1] = ti[j]; pos = j; }
                tv[pos] = score; ti[pos] = m4 + u;
            }
        }
    }
    int* op = idxb + (size_t)n * KK;
#pragma unroll
    for (int j = 0; j < KK; ++j) op[j] = ti[j];
}

// ---------------------------------------------------------------- gather + bn + relu + max_k
// out[b,n,o] = max_k relu( (P[b,idx[n,k],o] + Q[b,n,o]) * g[o]/sqrt(1+eps) + bias[o] )
__global__ void combine_kernel(const float* __restrict__ P, const float* __restrict__ Q,
                               const int* __restrict__ idx, const float* __restrict__ g,
                               const float* __restrict__ bv, float* __restrict__ XCcol,
                               int cout) {
    int bn = blockIdx.x;           // b*NN + n
    int o  = threadIdx.x;
    int b  = bn / NN;
    float scale = g[o] / sqrtf(1.f + BN_EPS);
    float bias  = bv[o];
    float q     = Q[(size_t)bn * cout + o];
    const int* ip = idx + (size_t)bn * KK;
    float acc = 0.f;                                    // relu >= 0
    for (int k = 0; k < KK; ++k) {
        int j   = ip[k];
        float h = (P[((size_t)b * NN + j) * cout + o] + q) * scale + bias;
        acc = fmaxf(acc, h);
    }
    XCcol[(size_t)bn * XC_LD + o] = acc;
}

// ---------------------------------------------------------------- conv5 epilogue: bn+relu+max_n
__global__ void reduce5_kernel(const float* __restrict__ H, const float* __restrict__ g,
                               const float* __restrict__ bv, float* __restrict__ hmax) {
    int t = blockIdx.x * blockDim.x + threadIdx.x;
    if (t >= BB * 1024) return;
    int b = t / 1024, o = t % 1024;
    float scale = g[o] / sqrtf(1.f + BN_EPS);
    float bias  = bv[o];
    const float* p = H + (size_t)b * NN * 1024 + o;
    float acc = 0.f;
    for (int n = 0; n < NN; ++n)
        acc = fmaxf(acc, p[(size_t)n * 1024] * scale + bias);
    hmax[t] = acc;
}

// ---------------------------------------------------------------- small FC (+ optional bn/relu)
__global__ void fc_kernel(const float* __restrict__ In, const float* __restrict__ W,
                          const float* __restrict__ g, const float* __restrict__ bv,
                          float* __restrict__ Out, int M, int Kd, int O, int do_relu) {
    int t = blockIdx.x * blockDim.x + threadIdx.x;
    if (t >= M * O) return;
    int m = t / O, o = t % O;
    const float* in = In + (size_t)m * Kd;
    const float* wr = W  + (size_t)o * Kd;
    float acc = 0.f;
    for (int k = 0; k < Kd; ++k) acc += in[k] * wr[k];
    if (g)       acc = acc * (g[o] / sqrtf(1.f + BN_EPS)) + bv[o];
    if (do_relu) acc = fmaxf(acc, 0.f);
    Out[t] = acc;
}

// ---------------------------------------------------------------- host-side layer driver
static void run_layer(const float* Xin, int lda, int cin, int cinp, int cout,
                      const float* W, const float* g, const float* bv,
                      float* XC, int colOff, float* sq, int* idx,
                      float* Wap, float* Wdp, float* P, float* Q, float* S,
                      hipStream_t s) {
    sq_kernel<<<(BB * NN + 255) / 256, 256, 0, s>>>(Xin, lda, cin, sq);
    for (int b = 0; b < BB; ++b) {
        gram_kernel<<<dim3(NN / 64, NN / 16), 32, 0, s>>>(
            Xin + (size_t)b * NN * lda, lda, cinp, sq + (size_t)b * NN, S);
        topk_kernel<<<NN / 128, 128, 0, s>>>(S, idx + (size_t)b * NN * KK);
    }
    wprep_kernel<<<(cout * cinp + 255) / 256, 256, 0, s>>>(W, cin, cinp, cout, Wap, Wdp);
    dim3 gg(cout / 64, (BB * NN) / 16);
    gemm_wmma_kernel<<<gg, 32, 0, s>>>(Xin, lda, Wap, cinp, P, cout, cinp);
    gemm_wmma_kernel<<<gg, 32, 0, s>>>(Xin, lda, Wdp, cinp, Q, cout, cinp);
    combine_kernel<<<BB * NN, cout, 0, s>>>(P, Q, idx, g, bv, XC + colOff, cout);
}

extern "C" void kernel_launch(void* const* d_in, const int* in_sizes, int n_in,
                              void* d_out, int out_size, void* d_ws, size_t ws_size,
                              hipStream_t stream) {
    const float* x   = (const float*)d_in[0];
    const float* w1  = (const float*)d_in[1];
    const float* w2  = (const float*)d_in[2];
    const float* w3  = (const float*)d_in[3];
    const float* w4  = (const float*)d_in[4];
    const float* w5  = (const float*)d_in[5];
    const float* fw1 = (const float*)d_in[6];
    const float* fw2 = (const float*)d_in[7];
    const float* fw3 = (const float*)d_in[8];
    const float* g1 = (const float*)d_in[9],  *b1 = (const float*)d_in[10];
    const float* g2 = (const float*)d_in[11], *b2 = (const float*)d_in[12];
    const float* g3 = (const float*)d_in[13], *b3 = (const float*)d_in[14];
    const float* g4 = (const float*)d_in[15], *b4 = (const float*)d_in[16];
    const float* g5 = (const float*)d_in[17], *b5 = (const float*)d_in[18];
    const float* g6 = (const float*)d_in[19], *b6 = (const float*)d_in[20];
    const float* g7 = (const float*)d_in[21], *b7 = (const float*)d_in[22];
    float* out = (float*)d_out;

    // ---- bump allocator over workspace (~103 MB; S/P/Q/H time-share one region) ----
    char* wp = (char*)d_ws;
    auto alloc = [&](size_t bytes) -> void* {
        void* p = wp;
        wp += (bytes + 255) & ~(size_t)255;
        return p;
    };
    float* XT0 = (float*)alloc((size_t)BB * NN * 4 * 4);         // padded c=4
    float* XC  = (float*)alloc((size_t)BB * NN * XC_LD * 4);     // [x1|x2|x3|x4] columns
    float* sq  = (float*)alloc((size_t)BB * NN * 4);
    int*   idx = (int*)  alloc((size_t)BB * NN * KK * 4);
    float* Wap = (float*)alloc((size_t)256 * 128 * 4);
    float* Wdp = (float*)alloc((size_t)256 * 128 * 4);
    char*  uni = (char*)alloc((size_t)BB * NN * 1024 * 4);       // 67 MB union region
    float* H   = (float*)uni;                                    // conv5 activations
    float* S   = (float*)uni;                                    // per-batch 2048x2048 scores
    float* P   = (float*)uni;                                    // edge-conv point GEMMs
    float* Q   = P + (size_t)BB * NN * 256;
    float* hmax = (float*)alloc((size_t)BB * 1024 * 4);
    float* h1   = (float*)alloc((size_t)BB * 512 * 4);
    float* h2   = (float*)alloc((size_t)BB * 256 * 4);

    // ---- x (B,3,N) -> point-major padded (B*N, 4) ----
    transpose_x_kernel<<<(BB * NN * 4 + 255) / 256, 256, 0, stream>>>(x, XT0);

    // ---- 4 edge-conv layers, outputs into XC column slices ----
    run_layer(XT0,      4,     3,   4,   64,  w1, g1, b1, XC, 0,   sq, idx, Wap, Wdp, P, Q, S, stream);
    run_layer(XC + 0,   XC_LD, 64,  64,  64,  w2, g2, b2, XC, 64,  sq, idx, Wap, Wdp, P, Q, S, stream);
    run_layer(XC + 64,  XC_LD, 64,  64,  128, w3, g3, b3, XC, 128, sq, idx, Wap, Wdp, P, Q, S, stream);
    run_layer(XC + 128, XC_LD, 128, 128, 256, w4, g4, b4, XC, 256, sq, idx, Wap, Wdp, P, Q, S, stream);

    // ---- conv5: (B*N,512) x (1024,512)^T -> H, then bn+relu+max over n ----
    {
        dim3 gg(1024 / 64, (BB * NN) / 16);
        gemm_wmma_kernel<<<gg, 32, 0, stream>>>(XC, XC_LD, w5, 512, H, 1024, 512);
        reduce5_kernel<<<(BB * 1024 + 255) / 256, 256, 0, stream>>>(H, g5, b5, hmax);
    }

    // ---- FC head ----
    fc_kernel<<<(BB * 512 + 255) / 256, 256, 0, stream>>>(hmax, fw1, g6, b6, h1, BB, 1024, 512, 1);
    fc_kernel<<<(BB * 256 + 255) / 256, 256, 0, stream>>>(h1,   fw2, g7, b7, h2, BB, 512, 256, 1);
    fc_kernel<<<(BB * 40 + 255) / 256, 256, 0, stream>>>(h2, fw3, nullptr, nullptr, out, BB, 256, 40, 0);
}